// TensorProductLayer_48576080118159
// MI455X (gfx1250) — compile-verified
//
#include <hip/hip_runtime.h>

// ---------------------------------------------------------------------------
// TensorProductLayer on MI455X (gfx1250).
// Bandwidth-bound (AI ~= 0.9 flop/byte; 800 MB total -> ~34us floor at
// 23.3 TB/s). No WMMA-suitable GEMM exists (only per-channel 3x3 algebra),
// so the CDNA5 feature exercised is the Tensor Data Mover: per 512-pair
// block, wave0 issues 6 TDM loads (global->LDS) + 3 TDM stores (LDS->global)
// as 1-D tile descriptors; all lanes compute from LDS with plain VALU FMAs.
// ---------------------------------------------------------------------------

#define PAIRS 512                 // (n,c) pairs per block, one per thread

typedef __attribute__((ext_vector_type(4))) unsigned int u32x4;
typedef __attribute__((ext_vector_type(4))) int          i32x4;
typedef __attribute__((ext_vector_type(8))) int          i32x8;

// Low 32 bits of a generic pointer into LDS == LDS byte address
// (flat aperture rule).
__device__ __forceinline__ unsigned lds_lo32(const void* p) {
    return (unsigned)(unsigned long long)(uintptr_t)p;
}

// Build a 1-D TDM descriptor moving `nd` dwords between global and LDS.
// Group0: [1:0]=count=1, [63:32]=lds_addr, [120:64]=global_addr, [127:126]=2.
// Group1: data_size=4B; tensor_dim0=tile_dim0=nd (exact OOB bound);
//         tensor_dim1=1; tile_dim1=tile_dim2=0 (unused); stride0=nd.
__device__ __forceinline__ void tdm_desc(u32x4& g0, i32x8& g1,
                                         const void* gptr,
                                         unsigned lds_bytes, unsigned nd) {
    unsigned long long ga = (unsigned long long)(uintptr_t)gptr;
    g0 = u32x4{ 1u,
                lds_bytes,
                (unsigned)(ga & 0xFFFFFFFFull),
                (unsigned)((ga >> 32) & 0x1FFFFFFull) | 0x80000000u };
    g1 = i32x8{ (int)0x00020000u,                          // data_size=2 (4B)
                (int)((nd & 0xFFFFu) << 16),               // tensor_dim0 lo
                (int)(((nd >> 16) & 0xFFFFu) | (1u << 16)),// td0 hi | td1 lo=1
                (int)((nd & 0xFFFFu) << 16),               // td1 hi=0 | tile_dim0
                0,                                         // tile_dim1/2 = 0
                (int)nd,                                   // stride0 lo
                0, 0 };                                    // stride0 hi, stride1
}

__device__ __forceinline__ void tdm_load(const void* gptr, unsigned lds_bytes,
                                         unsigned nd) {
    u32x4 g0; i32x8 g1;
    tdm_desc(g0, g1, gptr, lds_bytes, nd);
    i32x4 z4 = i32x4{0, 0, 0, 0};
#if defined(__clang_major__) && (__clang_major__ >= 23)
    i32x8 z8 = i32x8{0, 0, 0, 0, 0, 0, 0, 0};
    __builtin_amdgcn_tensor_load_to_lds(g0, g1, z4, z4, z8, 0);
#else
    __builtin_amdgcn_tensor_load_to_lds(g0, g1, z4, z4, 0);
#endif
}

__device__ __forceinline__ void tdm_store(void* gptr, unsigned lds_bytes,
                                          unsigned nd) {
    u32x4 g0; i32x8 g1;
    tdm_desc(g0, g1, gptr, lds_bytes, nd);
    i32x4 z4 = i32x4{0, 0, 0, 0};
#if defined(__clang_major__) && (__clang_major__ >= 23)
    i32x8 z8 = i32x8{0, 0, 0, 0, 0, 0, 0, 0};
    __builtin_amdgcn_tensor_store_from_lds(g0, g1, z4, z4, z8, 0);
#else
    __builtin_amdgcn_tensor_store_from_lds(g0, g1, z4, z4, 0);
#endif
}

// LDS layout (dwords): inputs then staged outputs. 19968 dwords = 78 KB.
#define OX0 0
#define OY0 (OX0 + PAIRS)          //  512
#define OX1 (OY0 + PAIRS)          // 1024
#define OY1 (OX1 + 3 * PAIRS)      // 2560
#define OX2 (OY1 + 3 * PAIRS)      // 4096
#define OY2 (OX2 + 9 * PAIRS)      // 8704
#define OO0 (OY2 + 9 * PAIRS)      // 13312
#define OO1 (OO0 + PAIRS)          // 13824
#define OO2 (OO1 + 3 * PAIRS)      // 15360
#define SMEM_DW (OO2 + 9 * PAIRS)  // 19968

__global__ __launch_bounds__(PAIRS)
void tp_layer_kernel(const float* __restrict__ x0, const float* __restrict__ x1,
                     const float* __restrict__ x2, const float* __restrict__ y0,
                     const float* __restrict__ y1, const float* __restrict__ y2,
                     float* __restrict__ out, int P) {
    __shared__ float sm[SMEM_DW];

    const unsigned base = (unsigned)blockIdx.x * PAIRS;   // first pair index

    // Wave 0 issues all input DMAs (TDM ignores EXEC; one issue per wave).
    if (threadIdx.x == 0) {
        unsigned sb = lds_lo32(sm);
        tdm_load(x0 + base,              sb + OX0 * 4u, PAIRS);
        tdm_load(y0 + base,              sb + OY0 * 4u, PAIRS);
        tdm_load(x1 + 3ull * base,       sb + OX1 * 4u, 3 * PAIRS);
        tdm_load(y1 + 3ull * base,       sb + OY1 * 4u, 3 * PAIRS);
        tdm_load(x2 + 9ull * base,       sb + OX2 * 4u, 9 * PAIRS);
        tdm_load(y2 + 9ull * base,       sb + OY2 * 4u, 9 * PAIRS);
        __builtin_amdgcn_s_wait_tensorcnt(0);
    }
    __syncthreads();

    const int t = threadIdx.x;
    const float a0 = sm[OX0 + t];
    const float b0 = sm[OY0 + t];
    float a1[3], b1[3], A[9], B[9];
#pragma unroll
    for (int i = 0; i < 3; ++i) {
        a1[i] = sm[OX1 + 3 * t + i];
        b1[i] = sm[OY1 + 3 * t + i];
    }
#pragma unroll
    for (int i = 0; i < 9; ++i) {
        A[i] = sm[OX2 + 9 * t + i];
        B[i] = sm[OY2 + 9 * t + i];
    }

    // out0: (0,0,0) + (1,1,0) dot + (2,2,0) Frobenius
    float o0 = a0 * b0 + a1[0] * b1[0] + a1[1] * b1[1] + a1[2] * b1[2];
#pragma unroll
    for (int k = 0; k < 9; ++k) o0 += A[k] * B[k];
    sm[OO0 + t] = o0;

    // out1[u]: (0,1,1) + (1,0,1) + (1,2,1) x1·y2 + (2,1,1) x2·y1
#pragma unroll
    for (int u = 0; u < 3; ++u) {
        float v = a0 * b1[u] + a1[u] * b0;
        v += a1[0] * B[u] + a1[1] * B[3 + u] + a1[2] * B[6 + u];
        v += A[3 * u] * b1[0] + A[3 * u + 1] * b1[1] + A[3 * u + 2] * b1[2];
        sm[OO1 + 3 * t + u] = v;
    }

    // out2[i,u]: (0,2,2) + (2,0,2) + (1,1,2) outer + (2,2,2) x2@y2
#pragma unroll
    for (int i = 0; i < 3; ++i) {
#pragma unroll
        for (int u = 0; u < 3; ++u) {
            float v = a0 * B[3 * i + u] + A[3 * i + u] * b0 + a1[i] * b1[u];
            v += A[3 * i] * B[u] + A[3 * i + 1] * B[3 + u] + A[3 * i + 2] * B[6 + u];
            sm[OO2 + 9 * t + 3 * i + u] = v;
        }
    }

    __syncthreads();   // all LDS output staging complete (s_wait_dscnt+barrier)

    // Wave 0 streams the staged tiles back to global via TDM stores.
    if (threadIdx.x == 0) {
        unsigned sb = lds_lo32(sm);
        tdm_store(out + base,                          sb + OO0 * 4u, PAIRS);
        tdm_store(out + (size_t)P + 3ull * base,       sb + OO1 * 4u, 3 * PAIRS);
        tdm_store(out + 4ull * (size_t)P + 9ull * base, sb + OO2 * 4u, 9 * PAIRS);
        __builtin_amdgcn_s_wait_tensorcnt(0);  // LDS must live until DMA drains
    }
}

extern "C" void kernel_launch(void* const* d_in, const int* in_sizes, int n_in,
                              void* d_out, int out_size, void* d_ws, size_t ws_size,
                              hipStream_t stream) {
    (void)n_in; (void)out_size; (void)d_ws; (void)ws_size;
    const float* x0 = (const float*)d_in[0];
    const float* x1 = (const float*)d_in[1];
    const float* x2 = (const float*)d_in[2];
    const float* y0 = (const float*)d_in[3];
    const float* y1 = (const float*)d_in[4];
    const float* y2 = (const float*)d_in[5];

    const int P = in_sizes[0];          // N*C = 5,120,000 (divisible by PAIRS)
    const int nblk = P / PAIRS;         // 10,000 blocks

    tp_layer_kernel<<<nblk, PAIRS, 0, stream>>>(x0, x1, x2, y0, y1, y2,
                                                (float*)d_out, P);
}